// LightRoIAlignExemplarEmbedder_24163486007436
// MI455X (gfx1250) — compile-verified
//
#include <hip/hip_runtime.h>
#include <hip/hip_bf16.h>

// ---------------------------------------------------------------------------
// LightRoIAlignExemplarEmbedder for MI455X (gfx1250, wave32)
//   z [16,128,64,64] f32, bboxes [16,64,4] f32, W_out [256,128] f32, b_out [256]
//   out [16,64,256] f32
// Phase 1: BCHW -> BHWC transpose (LDS tiled)        -> ws
// Phase 2: RoI-align 1x1 (adaptive grid, bilinear)   -> feats in ws
// Phase 3: feats @ W_out^T + b  via V_WMMA_F32_16X16X4_F32 (fp32-exact)
// ---------------------------------------------------------------------------

#define Bb   16
#define Nn   64
#define Cc   128
#define Hh   64
#define Ww   64
#define HW   (Hh * Ww)
#define OUTC 256
#define MAXG 16
#define ROI_SCALE 0.125f

typedef float v2f __attribute__((ext_vector_type(2)));
typedef float v8f __attribute__((ext_vector_type(8)));

// ---------------- Kernel 1: BCHW -> BHWC transpose (per batch: [C,HW] -> [HW,C])
__global__ void k_transpose_bchw_bhwc(const float* __restrict__ z,
                                      float* __restrict__ zf) {
    __shared__ float tile[32][33];               // +1 pad: no LDS bank conflicts
    const int bx = blockIdx.x;                   // HW/32 tiles
    const int by = blockIdx.y;                   // C/32 tiles
    const int bz = blockIdx.z;                   // batch
    const int tx = threadIdx.x;                  // 0..31
    const int ty = threadIdx.y;                  // 0..7

    const int p_in = bx * 32 + tx;               // pixel index (coalesced read)
#pragma unroll
    for (int j = 0; j < 4; ++j) {
        const int c_in = by * 32 + ty + j * 8;
        tile[ty + j * 8][tx] = z[((size_t)bz * Cc + c_in) * HW + p_in];
    }
    __syncthreads();

    const int c_out = by * 32 + tx;              // channel (coalesced write)
#pragma unroll
    for (int j = 0; j < 4; ++j) {
        const int p_out = bx * 32 + ty + j * 8;
        zf[((size_t)bz * HW + p_out) * Cc + c_out] = tile[tx][ty + j * 8];
    }
}

// torchvision bilinear edge handling (matches reference _prep_coord)
__device__ __forceinline__ void prep_coord(float c, int L,
                                           int& lo, int& hi, float& frac, bool& inr) {
    inr = (c > -1.0f) && (c < (float)L);
    float cc = fmaxf(c, 0.0f);
    lo = min((int)floorf(cc), L - 1);
    hi = min(lo + 1, L - 1);
    if (lo >= L - 1) cc = (float)lo;
    frac = cc - (float)lo;
}

// ---------------- Kernel 2: RoI-align 1x1. One block (8 waves) per ROI.
// Lane l owns channels [4l, 4l+4); each wave strides over grid points.
__global__ void k_roialign(const float* __restrict__ zf,        // [B,H,W,C]
                           const float* __restrict__ boxes,     // [B,N,4]
                           float* __restrict__ feats) {         // [B*N, C]
    __shared__ float red[8 * Cc];                // 4 KB cross-wave reduction

    const int roi  = blockIdx.x;                 // b*N + n
    const int b    = roi / Nn;
    const int tid  = threadIdx.x;
    const int lane = tid & 31;
    const int wave = tid >> 5;

    const float* bx = boxes + (size_t)roi * 4;
    const float x1 = bx[0] * ROI_SCALE - 0.5f;
    const float y1 = bx[1] * ROI_SCALE - 0.5f;
    const float x2 = bx[2] * ROI_SCALE - 0.5f;
    const float y2 = bx[3] * ROI_SCALE - 0.5f;
    const float roi_w = x2 - x1, roi_h = y2 - y1;
    const float gwf = fminf(fmaxf(ceilf(roi_w), 1.0f), (float)MAXG);
    const float ghf = fminf(fmaxf(ceilf(roi_h), 1.0f), (float)MAXG);
    const int gw = (int)gwf, gh = (int)ghf;
    const float bin_w = roi_w / gwf, bin_h = roi_h / ghf;

    const float4* zf4 = (const float4*)zf;       // 32 lanes x float4 == C=128
    const size_t base4 = (size_t)b * HW * 32;    // float4 units

    float a0 = 0.f, a1 = 0.f, a2 = 0.f, a3 = 0.f;
    const int npts = gh * gw;
    for (int p = wave; p < npts; p += 8) {
        const int gy = p / gw, gx = p - gy * gw;
        const float sy = y1 + ((float)gy + 0.5f) * bin_h;
        const float sx = x1 + ((float)gx + 0.5f) * bin_w;
        int yl, yh, xl, xh; float fy, fx; bool iry, irx;
        prep_coord(sy, Hh, yl, yh, fy, iry);
        prep_coord(sx, Ww, xl, xh, fx, irx);
        if (!(iry && irx)) continue;             // zero value, still counted
        const float hy = 1.0f - fy, hx = 1.0f - fx;
        const float w11 = hy * hx, w12 = hy * fx, w21 = fy * hx, w22 = fy * fx;
        const float4 v11 = zf4[base4 + ((size_t)yl * Ww + xl) * 32 + lane];
        const float4 v12 = zf4[base4 + ((size_t)yl * Ww + xh) * 32 + lane];
        const float4 v21 = zf4[base4 + ((size_t)yh * Ww + xl) * 32 + lane];
        const float4 v22 = zf4[base4 + ((size_t)yh * Ww + xh) * 32 + lane];
        a0 += w11 * v11.x + w12 * v12.x + w21 * v21.x + w22 * v22.x;
        a1 += w11 * v11.y + w12 * v12.y + w21 * v21.y + w22 * v22.y;
        a2 += w11 * v11.z + w12 * v12.z + w21 * v21.z + w22 * v22.z;
        a3 += w11 * v11.w + w12 * v12.w + w21 * v21.w + w22 * v22.w;
    }

    red[wave * Cc + lane * 4 + 0] = a0;
    red[wave * Cc + lane * 4 + 1] = a1;
    red[wave * Cc + lane * 4 + 2] = a2;
    red[wave * Cc + lane * 4 + 3] = a3;
    __syncthreads();

    if (tid < Cc) {
        float s = 0.f;
#pragma unroll
        for (int w = 0; w < 8; ++w) s += red[w * Cc + tid];
        feats[(size_t)roi * Cc + tid] = s / (ghf * gwf);
    }
}

// ---------------- Kernel 3: out[1024,256] = feats[1024,128] @ W^T + bias
// V_WMMA_F32_16X16X4_F32, one wave per 16x16 tile, 32 WMMAs over K=128.
__global__ void k_gemm_wmma(const float* __restrict__ feats,   // [1024,128]
                            const float* __restrict__ Wt,      // [256,128]
                            const float* __restrict__ bias,    // [256]
                            float* __restrict__ out) {         // [1024,256]
    const int tid  = threadIdx.x;
    const int lane = tid & 31;
    const int wave = tid >> 5;
    const int tileId = blockIdx.x * 8 + wave;    // 0..1023
    const int nt = tileId & 15;                  // N tile (fast): 8 waves share A rows
    const int mt = tileId >> 4;                  // M tile
    const int m0 = mt * 16, n0 = nt * 16;

    const int l    = lane & 15;                  // row (A) / col (B) within tile
    const int half = lane >> 4;                  // selects K pair {0,1} vs {2,3}

    const float* arow = feats + (size_t)(m0 + l) * Cc + half * 2;
    const float* bcol = Wt    + (size_t)(n0 + l) * Cc + half * 2;

    v8f acc = {};
#pragma unroll
    for (int kk = 0; kk < Cc; kk += 4) {
        // A 16x4: VGPR0/1 = K{0,1} (lanes 0-15) | K{2,3} (lanes 16-31)
        v2f a = *(const v2f*)(arow + kk);
        // B 4x16: mirrored striping; B[k][n] = W_out[n][k]
        v2f bm = *(const v2f*)(bcol + kk);
        acc = __builtin_amdgcn_wmma_f32_16x16x4_f32(
            false, a, false, bm, (short)0, acc, false, false);
    }

    const float bv = bias[n0 + l];
#pragma unroll
    for (int r = 0; r < 8; ++r) {
        // D layout: VGPR r -> M = m0 + r + 8*half, N = n0 + (lane&15)
        out[(size_t)(m0 + r + half * 8) * OUTC + (n0 + l)] = acc[r] + bv;
    }
}

extern "C" void kernel_launch(void* const* d_in, const int* in_sizes, int n_in,
                              void* d_out, int out_size, void* d_ws, size_t ws_size,
                              hipStream_t stream) {
    const float* z     = (const float*)d_in[0];  // [16,128,64,64]
    const float* boxes = (const float*)d_in[1];  // [16,64,4]
    const float* W_out = (const float*)d_in[2];  // [256,128]
    const float* b_out = (const float*)d_in[3];  // [256]
    float* out = (float*)d_out;                  // [16,64,256]

    float* zf    = (float*)d_ws;                                 // 32 MB
    float* feats = (float*)((char*)d_ws + (size_t)Bb * HW * Cc * 4); // 512 KB

    dim3 gT(HW / 32, Cc / 32, Bb);               // 128 x 4 x 16 tiles
    dim3 bT(32, 8);
    k_transpose_bchw_bhwc<<<gT, bT, 0, stream>>>(z, zf);

    k_roialign<<<Bb * Nn, 256, 0, stream>>>(zf, boxes, feats);

    // 1024 tiles of 16x16, 8 waves/block -> 128 blocks
    k_gemm_wmma<<<128, 256, 0, stream>>>(feats, W_out, b_out, out);
}